// PyGGAT_47777216200841
// MI455X (gfx1250) — compile-verified
//
#include <hip/hip_runtime.h>

#define C_IN 128
#define HC   128   // HEADS * C_HEAD
#define H    4
#define CH   32
#define NEG_SLOPE 0.2f

typedef __attribute__((ext_vector_type(2))) float v2f;
typedef __attribute__((ext_vector_type(8))) float v8f;

static __device__ __forceinline__ float leaky(float v) {
    return v > 0.f ? v : NEG_SLOPE * v;
}
// Order-preserving float -> uint key for atomicMax on floats (all finite values)
static __device__ __forceinline__ unsigned fkey(float f) {
    unsigned u = __float_as_uint(f);
    return (u & 0x80000000u) ? ~u : (u | 0x80000000u);
}
static __device__ __forceinline__ float funkey(unsigned u) {
    return (u & 0x80000000u) ? __uint_as_float(u & 0x7fffffffu) : __uint_as_float(~u);
}

// ---------------------------------------------------------------------------
// K1: xl = x @ W using V_WMMA_F32_16X16X4_F32. One wave per 16x16 output tile.
// A frag (16x4 f32): lane holds row M=lane&15, K = 2*(lane>>4)+r in reg r.
// B frag (4x16 f32): lane holds col N=lane&15, K = 2*(lane>>4)+r in reg r.
// C/D (16x16 f32): reg r holds M = r + 8*(lane>>4), N = lane&15.
// ---------------------------------------------------------------------------
__global__ __launch_bounds__(256) void gat_gemm_wmma(
    const float* __restrict__ x, const float* __restrict__ W,
    float* __restrict__ xl, int nNodes)
{
    const int wave = (blockIdx.x * blockDim.x + threadIdx.x) >> 5;
    const int lane = threadIdx.x & 31;
    const int tm = wave >> 3;   // 8 N-tiles of 16 (HC=128)
    const int tn = wave & 7;
    if (tm * 16 >= nNodes) return;   // wave-uniform: EXEC stays all-ones for WMMA

    const int hi   = lane >> 4;       // 0 or 1
    const int lidx = lane & 15;
    int row = tm * 16 + lidx;
    if (row >= nNodes) row = nNodes - 1;   // clamp load row (tail safety)
    const int col = tn * 16 + lidx;
    const int kk  = hi * 2;

    const float* __restrict__ xr = x + (size_t)row * C_IN;

    v8f acc = {0.f, 0.f, 0.f, 0.f, 0.f, 0.f, 0.f, 0.f};
    #pragma unroll
    for (int k0 = 0; k0 < C_IN; k0 += 4) {
        v2f a, b;
        a.x = xr[k0 + kk];
        a.y = xr[k0 + kk + 1];
        b.x = W[(size_t)(k0 + kk) * HC + col];
        b.y = W[(size_t)(k0 + kk + 1) * HC + col];
        acc = __builtin_amdgcn_wmma_f32_16x16x4_f32(
            /*neg_a=*/false, a, /*neg_b=*/false, b,
            /*c_mod=*/(short)0, acc, /*reuse_a=*/false, /*reuse_b=*/false);
    }

    #pragma unroll
    for (int r = 0; r < 8; ++r) {
        const int orow = tm * 16 + r + 8 * hi;
        if (orow < nNodes)
            xl[(size_t)orow * HC + col] = acc[r];
    }
}

// ---------------------------------------------------------------------------
// K2: per (node, head): a_src/a_dst dot products; init segment-max with the
// self-loop logit leaky(a_src[n]+a_dst[n]).
// ---------------------------------------------------------------------------
__global__ void gat_attn_node(
    const float* __restrict__ xl, const float* __restrict__ att_src,
    const float* __restrict__ att_dst, float* __restrict__ a_src,
    float* __restrict__ a_dst, unsigned* __restrict__ m_bits, int nNodes)
{
    const int t = blockIdx.x * blockDim.x + threadIdx.x;
    const int n = t >> 2, h = t & 3;
    if (n >= nNodes) return;
    const float* __restrict__ v  = xl + (size_t)n * HC + h * CH;
    const float* __restrict__ as = att_src + h * CH;
    const float* __restrict__ ad = att_dst + h * CH;
    float s = 0.f, d = 0.f;
    #pragma unroll
    for (int c = 0; c < CH; ++c) { const float xv = v[c]; s += xv * as[c]; d += xv * ad[c]; }
    a_src[n * H + h] = s;
    a_dst[n * H + h] = d;
    m_bits[n * H + h] = fkey(leaky(s + d));
}

// K3: segment max over edges (skip existing self loops)
__global__ void gat_edge_max(
    const long long* __restrict__ ei, const float* __restrict__ a_src,
    const float* __restrict__ a_dst, unsigned* __restrict__ m_bits, int nEdges)
{
    const int t = blockIdx.x * blockDim.x + threadIdx.x;
    const int e = t >> 2, h = t & 3;
    if (e >= nEdges) return;
    const long long s = ei[e], d = ei[(size_t)nEdges + e];
    if (s == d) return;
    const float ev = leaky(a_src[s * H + h] + a_dst[d * H + h]);
    atomicMax(&m_bits[d * H + h], fkey(ev));
}

// K4: decode max; init denom with self-loop exp term
__global__ void gat_node_denom_init(
    const float* __restrict__ a_src, const float* __restrict__ a_dst,
    const unsigned* __restrict__ m_bits, float* __restrict__ m_f,
    float* __restrict__ denom, int nNodes)
{
    const int t = blockIdx.x * blockDim.x + threadIdx.x;
    const int n = t >> 2, h = t & 3;
    if (n >= nNodes) return;
    const float m = funkey(m_bits[n * H + h]);
    m_f[n * H + h] = m;
    const float es = leaky(a_src[n * H + h] + a_dst[n * H + h]);
    denom[n * H + h] = __expf(es - m);
}

// K5: segment sum of exp over edges
__global__ void gat_edge_denom(
    const long long* __restrict__ ei, const float* __restrict__ a_src,
    const float* __restrict__ a_dst, const float* __restrict__ m_f,
    float* __restrict__ denom, int nEdges)
{
    const int t = blockIdx.x * blockDim.x + threadIdx.x;
    const int e = t >> 2, h = t & 3;
    if (e >= nEdges) return;
    const long long s = ei[e], d = ei[(size_t)nEdges + e];
    if (s == d) return;
    const float ev = leaky(a_src[s * H + h] + a_dst[d * H + h]);
    atomicAdd(&denom[d * H + h], __expf(ev - m_f[d * H + h]));
}

// K6: initialize out with the self-loop message (also clears 0xAA poison)
__global__ void gat_self_msg(
    const float* __restrict__ xl, const float* __restrict__ a_src,
    const float* __restrict__ a_dst, const float* __restrict__ m_f,
    const float* __restrict__ denom, float* __restrict__ out, int nNodes)
{
    const int t = blockIdx.x * blockDim.x + threadIdx.x;
    if (t >= nNodes * HC) return;
    const int n = t >> 7;          // / HC
    const int j = t & (HC - 1);
    const int h = j >> 5;          // / CH
    const float es = leaky(a_src[n * H + h] + a_dst[n * H + h]);
    const float alpha = __expf(es - m_f[n * H + h]) / denom[n * H + h];
    out[t] = xl[t] * alpha;
}

// K7: edge messages. One wave per edge; 4 heads x 32 lanes of atomic adds.
__global__ __launch_bounds__(256) void gat_edge_msg(
    const long long* __restrict__ ei, const float* __restrict__ xl,
    const float* __restrict__ a_src, const float* __restrict__ a_dst,
    const float* __restrict__ m_f, const float* __restrict__ denom,
    float* __restrict__ out, int nEdges)
{
    const int wave = (blockIdx.x * blockDim.x + threadIdx.x) >> 5;
    const int lane = threadIdx.x & 31;
    if (wave >= nEdges) return;
    const long long s = ei[wave], d = ei[(size_t)nEdges + wave];
    if (s == d) return;
    const float* __restrict__ xs = xl + (size_t)s * HC;
    float* __restrict__ od = out + (size_t)d * HC;
    #pragma unroll
    for (int h = 0; h < H; ++h) {
        const float ev = leaky(a_src[s * H + h] + a_dst[d * H + h]);
        const float alpha = __expf(ev - m_f[d * H + h]) / denom[d * H + h];
        atomicAdd(&od[h * CH + lane], xs[h * CH + lane] * alpha);
    }
}

// K8: out = (out + bias) / sqrt(H)
__global__ void gat_finalize(const float* __restrict__ bias,
                             float* __restrict__ out, int total)
{
    const int t = blockIdx.x * blockDim.x + threadIdx.x;
    if (t >= total) return;
    out[t] = (out[t] + bias[t & (HC - 1)]) * 0.5f;
}

extern "C" void kernel_launch(void* const* d_in, const int* in_sizes, int n_in,
                              void* d_out, int out_size, void* d_ws, size_t ws_size,
                              hipStream_t stream)
{
    const float*     x       = (const float*)d_in[0];
    const long long* ei      = (const long long*)d_in[1];
    const float*     W       = (const float*)d_in[2];
    const float*     att_src = (const float*)d_in[3];
    const float*     att_dst = (const float*)d_in[4];
    const float*     bias    = (const float*)d_in[5];
    float*           out     = (float*)d_out;

    const int nNodes = in_sizes[0] / C_IN;
    const int nEdges = in_sizes[1] / 2;

    // Workspace carve-up (256B aligned slices)
    char*  ws  = (char*)d_ws;
    size_t off = 0;
    auto carve = [&](size_t bytes) -> void* {
        void* p = ws + off;
        off = (off + bytes + 255) & ~(size_t)255;
        return p;
    };
    float*    xl     = (float*)   carve((size_t)nNodes * HC * sizeof(float));
    float*    a_src  = (float*)   carve((size_t)nNodes * H  * sizeof(float));
    float*    a_dst  = (float*)   carve((size_t)nNodes * H  * sizeof(float));
    unsigned* m_bits = (unsigned*)carve((size_t)nNodes * H  * sizeof(unsigned));
    float*    m_f    = (float*)   carve((size_t)nNodes * H  * sizeof(float));
    float*    denom  = (float*)   carve((size_t)nNodes * H  * sizeof(float));
    (void)ws_size;

    // K1: WMMA GEMM. waves = ceil(nNodes/16) * 8 N-tiles; 8 waves/block.
    {
        const int mtiles = (nNodes + 15) / 16;
        const int waves  = mtiles * 8;
        const int blocks = (waves + 7) / 8;
        gat_gemm_wmma<<<blocks, 256, 0, stream>>>(x, W, xl, nNodes);
    }
    // K2
    {
        const int threads = nNodes * H;
        gat_attn_node<<<(threads + 255) / 256, 256, 0, stream>>>(
            xl, att_src, att_dst, a_src, a_dst, m_bits, nNodes);
    }
    // K3
    {
        const int threads = nEdges * H;
        gat_edge_max<<<(threads + 255) / 256, 256, 0, stream>>>(
            ei, a_src, a_dst, m_bits, nEdges);
    }
    // K4
    {
        const int threads = nNodes * H;
        gat_node_denom_init<<<(threads + 255) / 256, 256, 0, stream>>>(
            a_src, a_dst, m_bits, m_f, denom, nNodes);
    }
    // K5
    {
        const int threads = nEdges * H;
        gat_edge_denom<<<(threads + 255) / 256, 256, 0, stream>>>(
            ei, a_src, a_dst, m_f, denom, nEdges);
    }
    // K6
    {
        const int threads = nNodes * HC;
        gat_self_msg<<<(threads + 255) / 256, 256, 0, stream>>>(
            xl, a_src, a_dst, m_f, denom, out, nNodes);
    }
    // K7: one wave per edge
    {
        const int blocks = (nEdges + 7) / 8;   // 8 waves (edges) per 256-thread block
        gat_edge_msg<<<blocks, 256, 0, stream>>>(
            ei, xl, a_src, a_dst, m_f, denom, out, nEdges);
    }
    // K8
    {
        const int total = nNodes * HC;
        gat_finalize<<<(total + 255) / 256, 256, 0, stream>>>(bias, out, total);
    }
}